// GCNModel_59004260713102
// MI455X (gfx1250) — compile-verified
//
#include <hip/hip_runtime.h>

#define N_NODES 100000
#define N_EDGES 1200000
#define DIM_IN  128
#define DIM_H   64
#define EPSV    1e-5f

typedef __attribute__((ext_vector_type(2))) float v2f;
typedef __attribute__((ext_vector_type(8))) float v8f;

// ---------------- degree / normalization ----------------

__global__ void k_init_deg(float* __restrict__ deg, int n) {
    int i = blockIdx.x * blockDim.x + threadIdx.x;
    if (i < n) deg[i] = 1.0f;  // self-loop contributes 1 to every node
}

__global__ void k_deg_accum(const long long* __restrict__ col, float* __restrict__ deg, int e) {
    int i = blockIdx.x * blockDim.x + threadIdx.x;
    if (i < e) atomicAdd(&deg[(size_t)col[i]], 1.0f);
}

__global__ void k_dinv(float* __restrict__ deg, int n) {
    int i = blockIdx.x * blockDim.x + threadIdx.x;
    if (i < n) {
        float d = deg[i];
        deg[i] = (d > 0.0f) ? rsqrtf(d) : 0.0f;
    }
}

// ---------------- WMMA GEMM: C[n x 64] = A[n x K] @ W[K x 64] ----------------
// One wave per 16-row tile; 4 accumulators cover all 64 output columns.
// V_WMMA_F32_16X16X4_F32 fragment layout (ISA 7.12.2):
//   A 16x4:  lanes 0-15 -> M=lane, {K=0,K=1}; lanes 16-31 -> M=lane-16, {K=2,K=3}
//   B 4x16:  lanes 0-15 -> N=lane, {K=0,K=1}; lanes 16-31 -> N=lane-16, {K=2,K=3}
//   C 16x16: VGPR r: lanes 0-15 -> M=r, N=lane; lanes 16-31 -> M=r+8, N=lane-16
//
// LDS holds W pair-interleaved so each B fragment is one contiguous v2f:
//   sW[(k>>1)*128 + 2*n + (k&1)] = W[k][n]   ->  single ds_load_b64 per fragment

__launch_bounds__(256)
__global__ void k_gemm_wmma(const float* __restrict__ A, const float* __restrict__ W,
                            float* __restrict__ C, int nrows, int K) {
    __shared__ float sW[DIM_IN * DIM_H];  // up to 32 KB (K=128)

    const int tid = threadIdx.x;
    const int kn  = K * DIM_H;
    for (int j = tid; j < kn; j += 256) {
        int p   = j >> 7;        // k-pair index
        int rem = j & 127;
        int n   = rem >> 1;
        int t   = rem & 1;
        sW[j] = W[((p << 1) + t) * DIM_H + n];
    }
    __syncthreads();

    const int wave = tid >> 5;
    const int lane = tid & 31;
    const int tile = blockIdx.x * 8 + wave;
    const int m0   = tile * 16;
    if (m0 >= nrows) return;                 // wave-uniform: EXEC stays all-ones

    const int lm   = lane & 15;              // row in A-frag / col in B-frag
    const int pofs = lane >> 4;              // which k-pair of the group of 4

    v8f c0 = {}, c1 = {}, c2 = {}, c3 = {};
    const float* arow = A + (size_t)(m0 + lm) * K;

    for (int k0 = 0; k0 < K; k0 += 4) {
        // A fragment: contiguous aligned pair {A[m][kk], A[m][kk+1]}
        const v2f a = *(const v2f*)(arow + k0 + (pofs << 1));
        // B fragments: contiguous v2f at pair index p, column n
        const v2f* bp = (const v2f*)&sW[((k0 >> 1) + pofs) << 7];
        c0 = __builtin_amdgcn_wmma_f32_16x16x4_f32(false, a, false, bp[lm +  0], (short)0, c0, false, false);
        c1 = __builtin_amdgcn_wmma_f32_16x16x4_f32(false, a, false, bp[lm + 16], (short)0, c1, false, false);
        c2 = __builtin_amdgcn_wmma_f32_16x16x4_f32(false, a, false, bp[lm + 32], (short)0, c2, false, false);
        c3 = __builtin_amdgcn_wmma_f32_16x16x4_f32(false, a, false, bp[lm + 48], (short)0, c3, false, false);
    }

    const int rbase = m0 + ((lane >> 4) << 3);  // +8 for upper half-wave
    for (int r = 0; r < 8; ++r) {
        float* cp = C + (size_t)(rbase + r) * DIM_H + lm;
        cp[0]  = c0[r];
        cp[16] = c1[r];
        cp[32] = c2[r];
        cp[48] = c3[r];
    }
}

// ---------------- self-loop init: acc[i][:] = dinv[i]^2 * hw[i][:] ----------------

__global__ void k_self_init(const float* __restrict__ hw, const float* __restrict__ dinv,
                            float* __restrict__ acc, int n) {
    int i = blockIdx.x * blockDim.x + threadIdx.x;
    if (i < n * DIM_H) {
        float di = dinv[i >> 6];
        acc[i] = di * di * hw[i];
    }
}

// ---------------- edge scatter: one wave per edge ----------------

__launch_bounds__(256)
__global__ void k_edge_scatter(const long long* __restrict__ row, const long long* __restrict__ col,
                               const float* __restrict__ dinv, const float* __restrict__ hw,
                               float* __restrict__ acc, int e) {
    int gw   = (blockIdx.x * blockDim.x + threadIdx.x) >> 5;
    int lane = threadIdx.x & 31;
    if (gw >= e) return;                     // wave-uniform guard
    long long r = row[gw];
    long long c = col[gw];
    float nrm = dinv[(size_t)r] * dinv[(size_t)c];
    const float* src = hw  + (size_t)r * DIM_H;
    float*       dst = acc + (size_t)c * DIM_H;
    atomicAdd(&dst[lane],      nrm * src[lane]);
    atomicAdd(&dst[lane + 32], nrm * src[lane + 32]);
}

// ---------------- bias + BatchNorm(eval) + ReLU ----------------

__global__ void k_bias_bn_relu(const float* __restrict__ acc,
                               const float* __restrict__ bias, const float* __restrict__ gamma,
                               const float* __restrict__ beta, const float* __restrict__ mean,
                               const float* __restrict__ var,
                               float* __restrict__ out, int n, int layer) {
    int i = blockIdx.x * blockDim.x + threadIdx.x;
    if (i < n * DIM_H) {
        int off = layer * DIM_H + (i & 63);
        float scale = gamma[off] * rsqrtf(var[off] + EPSV);
        float v = (acc[i] + bias[off] - mean[off]) * scale + beta[off];
        out[i] = (v > 0.0f) ? v : 0.0f;
    }
}

// ---------------- classifier: out[n x 2] = h @ cls_w + cls_b ----------------

__launch_bounds__(256)
__global__ void k_classifier(const float* __restrict__ h, const float* __restrict__ cw,
                             const float* __restrict__ cb, float* __restrict__ out, int n) {
    __shared__ float sw[DIM_H * 2];
    __shared__ float sb[2];
    if (threadIdx.x < DIM_H * 2) sw[threadIdx.x] = cw[threadIdx.x];
    if (threadIdx.x < 2)         sb[threadIdx.x] = cb[threadIdx.x];
    __syncthreads();
    int i = blockIdx.x * blockDim.x + threadIdx.x;
    if (i >= n) return;
    const float* hr = h + (size_t)i * DIM_H;
    float a0 = sb[0], a1 = sb[1];
    #pragma unroll
    for (int f = 0; f < DIM_H; ++f) {
        float v = hr[f];
        a0 += v * sw[2 * f];
        a1 += v * sw[2 * f + 1];
    }
    out[2 * i]     = a0;
    out[2 * i + 1] = a1;
}

// ---------------- driver ----------------

extern "C" void kernel_launch(void* const* d_in, const int* in_sizes, int n_in,
                              void* d_out, int out_size, void* d_ws, size_t ws_size,
                              hipStream_t stream) {
    const float*     x      = (const float*)d_in[0];
    const long long* ei     = (const long long*)d_in[1];  // int64 [2, E]
    const float*     w0     = (const float*)d_in[2];
    const float*     w1     = (const float*)d_in[3];
    const float*     w2     = (const float*)d_in[4];
    const float*     biases = (const float*)d_in[5];
    const float*     gamma  = (const float*)d_in[6];
    const float*     beta   = (const float*)d_in[7];
    const float*     rmean  = (const float*)d_in[8];
    const float*     rvar   = (const float*)d_in[9];
    const float*     cw     = (const float*)d_in[10];
    const float*     cb     = (const float*)d_in[11];
    float* out = (float*)d_out;

    const int n = N_NODES;
    const int e = N_EDGES;
    const long long* row  = ei;        // sources
    const long long* colv = ei + e;    // targets

    float* dinv = (float*)d_ws;
    float* hw   = dinv + ((n + 255) & ~255);
    float* acc  = hw  + (size_t)n * DIM_H;
    float* hbuf = acc + (size_t)n * DIM_H;

    dim3 b256(256);

    // symmetric normalization coefficients
    k_init_deg <<<(n + 255) / 256, b256, 0, stream>>>(dinv, n);
    k_deg_accum<<<(e + 255) / 256, b256, 0, stream>>>(colv, dinv, e);
    k_dinv     <<<(n + 255) / 256, b256, 0, stream>>>(dinv, n);

    const float* wl[3] = { w0, w1, w2 };
    const float* hin = x;
    const int tiles = n / 16;  // 6250, exact
    for (int l = 0; l < 3; ++l) {
        int K = (l == 0) ? DIM_IN : DIM_H;
        k_gemm_wmma   <<<(tiles + 7) / 8, b256, 0, stream>>>(hin, wl[l], hw, n, K);
        k_self_init   <<<(n * DIM_H + 255) / 256, b256, 0, stream>>>(hw, dinv, acc, n);
        k_edge_scatter<<<(e * 32 + 255) / 256, b256, 0, stream>>>(row, colv, dinv, hw, acc, e);
        k_bias_bn_relu<<<(n * DIM_H + 255) / 256, b256, 0, stream>>>(acc, biases, gamma, beta,
                                                                     rmean, rvar, hbuf, n, l);
        hin = hbuf;
    }
    k_classifier<<<(n + 255) / 256, b256, 0, stream>>>(hbuf, cw, cb, out, n);
}